// GNNML3_64991445123448
// MI455X (gfx1250) — compile-verified
//
#include <hip/hip_runtime.h>

typedef __attribute__((ext_vector_type(2))) float v2f;
typedef __attribute__((ext_vector_type(8))) float v8f;

#define NNODES 50000
#define NEDGES 1600000
#define NFREQ  10
#define CO     32            // conv output channels
#define GO     16            // gate output channels
#define CFULL  48            // concat width
#define PW     (NFREQ * CO)  // 320, width of P
#define MT     (NNODES / 16) // 3125 exact M-tiles

// acc[i,o] = b[o]
__global__ __launch_bounds__(256) void init_acc_kernel(float* __restrict__ acc,
                                                       const float* __restrict__ b) {
  int i = blockIdx.x * blockDim.x + threadIdx.x;
  if (i < NNODES * CO) acc[i] = b[i & (CO - 1)];
}

// ---------------- Layer 1 (C = 1): rank-1 outer products, no GEMM needed -------

// P[i, col] = x[i] * W1[col]   (W1 is [10,1,32] = flat [320])
__global__ __launch_bounds__(256) void p1_kernel(const float* __restrict__ x,
                                                 const float* __restrict__ W,
                                                 float* __restrict__ P) {
  int i = blockIdx.x * blockDim.x + threadIdx.x;
  if (i < NNODES * PW) {
    int node = i / PW;
    int col  = i - node * PW;
    P[i] = x[node] * W[col];
  }
}

// xo[i, 32+n] = (x[i]*U[n] + ub[n]) * (x[i]*V[n] + vb[n])   (U,V flat [16])
__global__ __launch_bounds__(256) void gate1_kernel(const float* __restrict__ x,
                                                    const float* __restrict__ U,
                                                    const float* __restrict__ ub,
                                                    const float* __restrict__ V,
                                                    const float* __restrict__ vb,
                                                    float* __restrict__ xo) {
  int i = blockIdx.x * blockDim.x + threadIdx.x;
  if (i < NNODES * GO) {
    int node = i >> 4;
    int n    = i & 15;
    float xv = x[node];
    xo[(size_t)node * CFULL + CO + n] = (xv * U[n] + ub[n]) * (xv * V[n] + vb[n]);
  }
}

// ---------------- Layers 2/3 (C = 48): f32 WMMA GEMMs, fully unrolled ----------

// P[i, k*32+o] = sum_c x[i,c] * W[k,c,o]   ([N x 48] @ [48 x 320])
__global__ __launch_bounds__(128) void wmma_P48_kernel(const float* __restrict__ x,
                                                       const float* __restrict__ W,
                                                       float* __restrict__ P) {
  const int lane  = threadIdx.x;                  // 0..31
  const int tileM = blockIdx.x;                   // 0..3124
  const int tileN = blockIdx.y * 4 + threadIdx.y; // 0..19
  const int half  = lane >> 4;
  const int l15   = lane & 15;
  const int row   = tileM * 16 + l15;             // A row
  const int ncol  = tileN * 16 + l15;             // B/D column
  const int kd    = ncol >> 5;                    // spectral support index
  const int o     = ncol & 31;                    // output channel

  const float* xr = x + (size_t)row * CFULL + half * 2;            // adjacent pair
  const float* Wk = W + (size_t)kd * CFULL * CO + o + (size_t)(half * 2) * CO;

  v8f acc = {};
#pragma unroll
  for (int kk = 0; kk < CFULL; kk += 4) {
    v2f a, b;
    a.x = xr[kk];
    a.y = xr[kk + 1];                 // merges into b64 load
    b.x = Wk[(size_t)kk * CO];
    b.y = Wk[(size_t)(kk + 1) * CO];
    acc = __builtin_amdgcn_wmma_f32_16x16x4_f32(false, a, false, b, (short)0, acc,
                                                false, false);
  }
  float* Pp = P + ((size_t)(tileM * 16 + half * 8)) * PW + ncol;
#pragma unroll
  for (int r = 0; r < 8; ++r) Pp[(size_t)r * PW] = acc[r];
}

// xo[i, 32+n] = (x@U + ub)[i,n] * (x@V + vb)[i,n]   (two WMMA accumulators)
__global__ __launch_bounds__(128) void wmma_gate48_kernel(const float* __restrict__ x,
                                                          const float* __restrict__ U,
                                                          const float* __restrict__ ub,
                                                          const float* __restrict__ V,
                                                          const float* __restrict__ vb,
                                                          float* __restrict__ xo) {
  int tileM = blockIdx.x * 4 + threadIdx.y;
  if (tileM >= MT) return;
  const int lane = threadIdx.x;
  const int half = lane >> 4;
  const int l15  = lane & 15;
  const int row  = tileM * 16 + l15;

  const float* xr = x + (size_t)row * CFULL + half * 2;
  const float* Up = U + l15 + (size_t)(half * 2) * GO;
  const float* Vp = V + l15 + (size_t)(half * 2) * GO;

  v8f cu = {}, cv = {};
#pragma unroll
  for (int kk = 0; kk < CFULL; kk += 4) {
    v2f a, bu, bv;
    a.x  = xr[kk];
    a.y  = xr[kk + 1];
    bu.x = Up[(size_t)kk * GO];
    bu.y = Up[(size_t)(kk + 1) * GO];
    bv.x = Vp[(size_t)kk * GO];
    bv.y = Vp[(size_t)(kk + 1) * GO];
    cu = __builtin_amdgcn_wmma_f32_16x16x4_f32(false, a, false, bu, (short)0, cu,
                                               false, false);
    cv = __builtin_amdgcn_wmma_f32_16x16x4_f32(false, a, false, bv, (short)0, cv,
                                               false, false);
  }
  float ubn = ub[l15], vbn = vb[l15];
#pragma unroll
  for (int r = 0; r < 8; ++r) {
    int m = tileM * 16 + r + half * 8;
    xo[(size_t)m * CFULL + CO + l15] = (cu[r] + ubn) * (cv[r] + vbn);
  }
}

// ---------------- Edge scatter stage (bandwidth/atomic bound) ------------------

// per edge: y[o] = sum_k ea[e,k] * P[src[e], k*32+o]; atomic scatter to acc[dst[e]]
__global__ __launch_bounds__(256) void edge_kernel(const float* __restrict__ ea,
                                                   const int* __restrict__ src,
                                                   const int* __restrict__ dst,
                                                   const float* __restrict__ P,
                                                   float* __restrict__ acc) {
  int e = blockIdx.x * 8 + threadIdx.y;
  if (e >= NEDGES) return;
  int o = threadIdx.x;  // 0..31 = output channel
  int s = src[e];
  int d = dst[e];
  const float* Ps  = P + (size_t)s * PW + o;
  const float* eap = ea + (size_t)e * NFREQ;
  float eav = (o < NFREQ) ? eap[o] : 0.f;  // one coalesced load, broadcast via shfl
  float y = 0.f;
#pragma unroll
  for (int k = 0; k < NFREQ; ++k) {
    y += __shfl(eav, k, 32) * Ps[(size_t)k * CO];
  }
  unsafeAtomicAdd(&acc[(size_t)d * CO + o], y);
}

// xo[i, 0:32] = relu(acc[i, :])
__global__ __launch_bounds__(256) void relu_kernel(const float* __restrict__ acc,
                                                   float* __restrict__ xo) {
  int i = blockIdx.x * blockDim.x + threadIdx.x;
  if (i < NNODES * CO) {
    int node = i >> 5;
    int o    = i & 31;
    float v  = acc[i];
    xo[(size_t)node * CFULL + o] = v > 0.f ? v : 0.f;
  }
}

// out[i] = fc2_b + x3[i,:] . fc2_w
__global__ __launch_bounds__(256) void fc2_kernel(const float* __restrict__ x,
                                                  const float* __restrict__ w,
                                                  const float* __restrict__ b,
                                                  float* __restrict__ out) {
  int i = blockIdx.x * blockDim.x + threadIdx.x;
  if (i < NNODES) {
    float s = b[0];
#pragma unroll
    for (int c = 0; c < CFULL; ++c) s += x[(size_t)i * CFULL + c] * w[c];
    out[i] = s;
  }
}

extern "C" void kernel_launch(void* const* d_in, const int* in_sizes, int n_in,
                              void* d_out, int out_size, void* d_ws, size_t ws_size,
                              hipStream_t stream) {
  const float* x0  = (const float*)d_in[0];
  const float* ea  = (const float*)d_in[1];
  const int*   ei  = (const int*)d_in[2];
  const int*   src = ei;           // row 0 = source j
  const int*   dst = ei + NEDGES;  // row 1 = target i

  const float* fc2w = (const float*)d_in[21];
  const float* fc2b = (const float*)d_in[22];

  float* ws  = (float*)d_ws;
  float* P   = ws;                              // [N, 320]
  float* acc = P + (size_t)NNODES * PW;         // [N, 32]
  float* xa  = acc + (size_t)NNODES * CO;       // [N, 48]
  float* xb  = xa + (size_t)NNODES * CFULL;     // [N, 48]

  const dim3 b256(256);
  const dim3 gNCO((NNODES * CO + 255) / 256);

  // ---------------- layer 1 (C=1) ----------------
  {
    const float* W  = (const float*)d_in[3];
    const float* b  = (const float*)d_in[4];
    const float* U  = (const float*)d_in[5];
    const float* ub = (const float*)d_in[6];
    const float* V  = (const float*)d_in[7];
    const float* vb = (const float*)d_in[8];
    init_acc_kernel<<<gNCO, b256, 0, stream>>>(acc, b);
    p1_kernel<<<dim3((NNODES * PW + 255) / 256), b256, 0, stream>>>(x0, W, P);
    edge_kernel<<<dim3(NEDGES / 8), dim3(32, 8), 0, stream>>>(ea, src, dst, P, acc);
    gate1_kernel<<<dim3((NNODES * GO + 255) / 256), b256, 0, stream>>>(x0, U, ub, V,
                                                                       vb, xa);
    relu_kernel<<<gNCO, b256, 0, stream>>>(acc, xa);
  }

  // ---------------- layers 2/3 (C=48, WMMA) ----------------
  auto run_layer48 = [&](const float* xin, int pbase, float* xout) {
    const float* W  = (const float*)d_in[pbase + 0];
    const float* b  = (const float*)d_in[pbase + 1];
    const float* U  = (const float*)d_in[pbase + 2];
    const float* ub = (const float*)d_in[pbase + 3];
    const float* V  = (const float*)d_in[pbase + 4];
    const float* vb = (const float*)d_in[pbase + 5];
    init_acc_kernel<<<gNCO, b256, 0, stream>>>(acc, b);
    wmma_P48_kernel<<<dim3(MT, 5), dim3(32, 4), 0, stream>>>(xin, W, P);
    edge_kernel<<<dim3(NEDGES / 8), dim3(32, 8), 0, stream>>>(ea, src, dst, P, acc);
    wmma_gate48_kernel<<<dim3((MT + 3) / 4), dim3(32, 4), 0, stream>>>(xin, U, ub, V,
                                                                       vb, xout);
    relu_kernel<<<gNCO, b256, 0, stream>>>(acc, xout);
  };

  run_layer48(xa, 9, xb);   // layer 2
  run_layer48(xb, 15, xa);  // layer 3

  fc2_kernel<<<dim3((NNODES + 255) / 256), b256, 0, stream>>>(xa, fc2w, fc2b,
                                                              (float*)d_out);
}